// GNNClassifier_57080115364431
// MI455X (gfx1250) — compile-verified
//
#include <hip/hip_runtime.h>
#include <math.h>

// ---------------------------------------------------------------------------
// GAT GNN classifier for gfx1250 (MI455X).
// Dense feature GEMMs run on the CDNA5 matrix pipe via V_WMMA_F32_16X16X4_F32
// (full f32 precision, matching the f32 reference; the GEMMs here are
// L2-bandwidth bound at ~23 GFLOP total, so low-precision WMMA buys nothing).
// Each wave register-blocks a 16x64 C slab: one A fragment feeds 4 independent
// WMMA accumulator chains (4x less X traffic, XDL latency hidden by ILP).
// Edge softmax/aggregation uses native float atomics (GLOBAL_ATOMIC_*_F32).
// ---------------------------------------------------------------------------

typedef __attribute__((ext_vector_type(2))) float v2f;
typedef __attribute__((ext_vector_type(8))) float v8f;

// ---------------- generic fill ----------------
__global__ void __launch_bounds__(256)
fill_kernel(float* __restrict__ p, float v, long long n) {
  long long i = (long long)blockIdx.x * blockDim.x + threadIdx.x;
  long long stride = (long long)gridDim.x * blockDim.x;
  for (; i < n; i += stride) p[i] = v;
}

// ---------------- f32 WMMA GEMM: C[M,Nc] = X[M,K] @ W[K,Nc] ----------------
// One wave per 16x64 C slab (4 x 16x16 WMMA subtiles sharing the A fragment).
// A-frag: lane r=lane&15 holds row m0+r, K = kk,kk+1 (kk=(lane>>4)*2) in
// v[0],v[1]. B-frag mirrors (rows K=kk,kk+1 striped across lanes).
// C/D: VGPR g -> row m0 + g + 8*(lane>=16), col n0+(lane&15).
__global__ void __launch_bounds__(256)
gemm_f32_wmma(const float* __restrict__ X, const float* __restrict__ W,
              float* __restrict__ C, int M, int K, int Nc) {
  const int lane = threadIdx.x;              // 0..31 (wave32)
  const int tilesN = (Nc + 63) >> 6;         // 64-wide N slabs
  const int tilesM = (M + 15) >> 4;
  const int tile = blockIdx.x * 8 + threadIdx.y;
  if (tile >= tilesM * tilesN) return;
  const int tn = tile % tilesN;              // fastest over N: waves of a
  const int tm = tile / tilesN;              // block share the same X rows
  const int m0 = tm << 4;
  const int n0 = tn << 6;
  const int r  = lane & 15;
  const int kk = (lane >> 4) << 1;           // 0 or 2

  int mr = m0 + r; if (mr > M - 1) mr = M - 1;   // clamp tail rows (loads only)
  const float* __restrict__ xrow = X + (size_t)mr * K + kk;
  const float* __restrict__ wb   = W + (size_t)kk * Nc + n0 + r;

  // wave-uniform subtile validity (Nc=512/768 are multiples of 64, so these
  // are all-true in practice; kept for generality)
  const bool v1 = (n0 + 16 + 15) < Nc;
  const bool v2 = (n0 + 32 + 15) < Nc;
  const bool v3 = (n0 + 48 + 15) < Nc;
  const int c1 = v1 ? 16 : 0;                // clamp invalid loads to col 0
  const int c2 = v2 ? 32 : 0;
  const int c3 = v3 ? 48 : 0;

  v8f a0 = {}, a1 = {}, a2 = {}, a3 = {};
  #pragma unroll 2
  for (int k0 = 0; k0 < K; k0 += 4) {
    v2f a = *(const v2f*)(xrow + k0);        // K,K+1 contiguous -> b64 load
    const float* w0 = wb + (size_t)k0 * Nc;  // row K=kk   of this K block
    const float* w1 = w0 + Nc;               // row K=kk+1
    v2f b0, b1, b2, b3;
    b0.x = w0[0];  b0.y = w1[0];
    b1.x = w0[c1]; b1.y = w1[c1];
    b2.x = w0[c2]; b2.y = w1[c2];
    b3.x = w0[c3]; b3.y = w1[c3];
    a0 = __builtin_amdgcn_wmma_f32_16x16x4_f32(false, a, false, b0, (short)0, a0, false, false);
    a1 = __builtin_amdgcn_wmma_f32_16x16x4_f32(false, a, false, b1, (short)0, a1, false, false);
    a2 = __builtin_amdgcn_wmma_f32_16x16x4_f32(false, a, false, b2, (short)0, a2, false, false);
    a3 = __builtin_amdgcn_wmma_f32_16x16x4_f32(false, a, false, b3, (short)0, a3, false, false);
  }

  const int rowbase = m0 + ((lane >> 4) << 3);
  float* cp = C + (size_t)rowbase * Nc + n0 + r;
  #pragma unroll
  for (int g = 0; g < 8; ++g) {
    if (rowbase + g < M) {
      float* row = cp + (size_t)g * Nc;
      row[0] = a0[g];
      if (v1) row[16] = a1[g];
      if (v2) row[32] = a2[g];
      if (v3) row[48] = a3[g];
    }
  }
}

// ---------------- attention scores: a_s[n,h], a_d[n,h] ----------------
// One wave per (node, head); coalesced loads + shuffle reduction.
__global__ void __launch_bounds__(256)
attn_scores(const float* __restrict__ Hm, const float* __restrict__ As,
            const float* __restrict__ Ad, float* __restrict__ as_out,
            float* __restrict__ ad_out, int Nn, int Hh, int Cc) {
  const int lane = threadIdx.x;
  const int pair = blockIdx.x * 8 + threadIdx.y;
  if (pair >= Nn * Hh) return;
  const int n = pair / Hh, h = pair % Hh;
  const float* hp = Hm + ((size_t)n * Hh + h) * Cc;
  const float* sp = As + (size_t)h * Cc;
  const float* dp = Ad + (size_t)h * Cc;
  float s = 0.f, d = 0.f;
  for (int c = lane; c < Cc; c += 32) {
    float hv = hp[c];
    s += hv * sp[c];
    d += hv * dp[c];
  }
  for (int off = 16; off > 0; off >>= 1) {
    s += __shfl_xor(s, off, 32);
    d += __shfl_xor(d, off, 32);
  }
  if (lane == 0) { as_out[pair] = s; ad_out[pair] = d; }
}

// ---------------- float atomic max (sign-split punning) ----------------
__device__ __forceinline__ void atomicMaxF(float* addr, float v) {
  if (v >= 0.f) atomicMax((int*)addr, __float_as_int(v));
  else          atomicMin((unsigned int*)addr, __float_as_uint(v));
}

__device__ __forceinline__ float leaky02(float e) {
  return e > 0.f ? e : 0.2f * e;
}

// ---------------- edge pass 1: segment max of leaky_relu scores ----------------
__global__ void __launch_bounds__(256)
edge_max(const int* __restrict__ ei, const float* __restrict__ as_,
         const float* __restrict__ ad_, float* __restrict__ m,
         int E, int Nn, int Hh) {
  long long t = (long long)blockIdx.x * blockDim.x + threadIdx.x;
  long long total = (long long)(E + Nn) * Hh;
  if (t >= total) return;
  const int e = (int)(t / Hh), h = (int)(t % Hh);
  const int s = (e < E) ? ei[e]     : (e - E);   // self-loops appended
  const int d = (e < E) ? ei[E + e] : (e - E);
  const float ev = leaky02(as_[(size_t)s * Hh + h] + ad_[(size_t)d * Hh + h]);
  atomicMaxF(m + (size_t)d * Hh + h, ev);
}

// ---------------- edge pass 2: segment sum of exp(e - m[dst]) ----------------
__global__ void __launch_bounds__(256)
edge_expsum(const int* __restrict__ ei, const float* __restrict__ as_,
            const float* __restrict__ ad_, const float* __restrict__ m,
            float* __restrict__ den, int E, int Nn, int Hh) {
  long long t = (long long)blockIdx.x * blockDim.x + threadIdx.x;
  long long total = (long long)(E + Nn) * Hh;
  if (t >= total) return;
  const int e = (int)(t / Hh), h = (int)(t % Hh);
  const int s = (e < E) ? ei[e]     : (e - E);
  const int d = (e < E) ? ei[E + e] : (e - E);
  const float ev = leaky02(as_[(size_t)s * Hh + h] + ad_[(size_t)d * Hh + h]);
  atomicAdd(den + (size_t)d * Hh + h, __expf(ev - m[(size_t)d * Hh + h]));
}

// ---------------- edge pass 3: out[dst,h,:] += alpha * h[src,h,:] ----------------
// One wave per edge; lanes stride the 128 channels (coalesced atomics).
__global__ void __launch_bounds__(256)
edge_agg(const int* __restrict__ ei, const float* __restrict__ as_,
         const float* __restrict__ ad_, const float* __restrict__ m,
         const float* __restrict__ den, const float* __restrict__ Hm,
         float* __restrict__ Om, int E, int Nn, int Hh, int Cc) {
  const int lane = threadIdx.x;
  const int e = blockIdx.x * 8 + threadIdx.y;
  if (e >= E + Nn) return;
  const int s = (e < E) ? ei[e]     : (e - E);
  const int d = (e < E) ? ei[E + e] : (e - E);
  for (int h = 0; h < Hh; ++h) {
    const float ev = leaky02(as_[(size_t)s * Hh + h] + ad_[(size_t)d * Hh + h]);
    const float alpha = __expf(ev - m[(size_t)d * Hh + h]) / den[(size_t)d * Hh + h];
    const float* hp = Hm + ((size_t)s * Hh + h) * Cc;
    float* op = Om + ((size_t)d * Hh + h) * Cc;
    for (int c = lane; c < Cc; c += 32)
      atomicAdd(op + c, alpha * hp[c]);
  }
}

// ---------------- head mean + bias + relu ----------------
__global__ void __launch_bounds__(256)
head_mean_relu(const float* __restrict__ Om, const float* __restrict__ bias,
               float* __restrict__ Xo, int Nn, int Hh, int Cc) {
  long long t = (long long)blockIdx.x * blockDim.x + threadIdx.x;
  if (t >= (long long)Nn * Cc) return;
  const int n = (int)(t / Cc), c = (int)(t % Cc);
  float acc = 0.f;
  for (int h = 0; h < Hh; ++h)
    acc += Om[((size_t)n * Hh + h) * Cc + c];
  const float v = acc / (float)Hh + bias[c];
  Xo[t] = v > 0.f ? v : 0.f;
}

// ---------------- global max pool per graph ----------------
__global__ void __launch_bounds__(256)
pool_max(const float* __restrict__ X, const int* __restrict__ batch,
         float* __restrict__ P, int Nn, int Cc) {
  long long t = (long long)blockIdx.x * blockDim.x + threadIdx.x;
  if (t >= (long long)Nn * Cc) return;
  const int n = (int)(t / Cc), c = (int)(t % Cc);
  atomicMaxF(P + (size_t)batch[n] * Cc + c, X[t]);
}

// ---------------- final linear head (one wave per graph) ----------------
__global__ void __launch_bounds__(256)
final_linear(const float* __restrict__ Pool, const float* __restrict__ w,
             const float* __restrict__ b, float* __restrict__ out,
             int Gg, int Cc) {
  const int lane = threadIdx.x;
  const int g = blockIdx.x * 8 + threadIdx.y;
  if (g >= Gg) return;
  float s = 0.f;
  for (int c = lane; c < Cc; c += 32)
    s += Pool[(size_t)g * Cc + c] * w[c];
  for (int off = 16; off > 0; off >>= 1)
    s += __shfl_xor(s, off, 32);
  if (lane == 0) out[g] = s + b[0];
}

// ---------------------------------------------------------------------------
extern "C" void kernel_launch(void* const* d_in, const int* in_sizes, int n_in,
                              void* d_out, int out_size, void* d_ws, size_t ws_size,
                              hipStream_t stream) {
  const float* x    = (const float*)d_in[0];
  const int*   ei   = (const int*)d_in[1];
  const int*   bat  = (const int*)d_in[2];
  const float* W1   = (const float*)d_in[3];
  const float* as1  = (const float*)d_in[4];
  const float* ad1  = (const float*)d_in[5];
  const float* b1   = (const float*)d_in[6];
  const float* W2   = (const float*)d_in[7];
  const float* as2  = (const float*)d_in[8];
  const float* ad2  = (const float*)d_in[9];
  const float* b2   = (const float*)d_in[10];
  const float* linW = (const float*)d_in[11];
  const float* linb = (const float*)d_in[12];
  float* out = (float*)d_out;

  const int Nn  = in_sizes[2];            // 50000 (batch vector length)
  const int E   = in_sizes[1] / 2;        // 400000
  const int Cc  = in_sizes[6];            // 128 (HID, from b1)
  const int H1h = in_sizes[4] / Cc;       // 4
  const int H2h = in_sizes[8] / Cc;       // 6
  const int IN  = in_sizes[0] / Nn;       // 256
  const int Gg  = out_size;               // 128
  const int Hmax = (H1h > H2h) ? H1h : H2h;
  (void)n_in; (void)ws_size;

  // workspace carve-up (reused across both layers)
  char* ws = (char*)d_ws;
  size_t off = 0;
  auto alloc = [&](size_t bytes) -> void* {
    void* p = ws + off;
    off += (bytes + 255) & ~(size_t)255;
    return p;
  };
  float* bufH = (float*)alloc((size_t)Nn * Hmax * Cc * sizeof(float)); // h = x@W
  float* bufO = (float*)alloc((size_t)Nn * Hmax * Cc * sizeof(float)); // aggregated
  float* bufX = (float*)alloc((size_t)Nn * Cc * sizeof(float));        // layer io
  float* aS   = (float*)alloc((size_t)Nn * Hmax * sizeof(float));
  float* aD   = (float*)alloc((size_t)Nn * Hmax * sizeof(float));
  float* mM   = (float*)alloc((size_t)Nn * Hmax * sizeof(float));
  float* den  = (float*)alloc((size_t)Nn * Hmax * sizeof(float));
  float* pool = (float*)alloc((size_t)Gg * Cc * sizeof(float));

  auto run_layer = [&](const float* Xin, int Kin, const float* Wm,
                       const float* attS, const float* attD, const float* bias,
                       int Hh, float* Xout) {
    const int Nout = Hh * Cc;
    // dense GEMM on the matrix pipe (16x64 slab per wave)
    {
      const int tiles = ((Nn + 15) / 16) * ((Nout + 63) / 64);
      gemm_f32_wmma<<<dim3((tiles + 7) / 8), dim3(32, 8), 0, stream>>>(
          Xin, Wm, bufH, Nn, Kin, Nout);
    }
    // per-(node,head) attention dots
    {
      const int pairs = Nn * Hh;
      attn_scores<<<dim3((pairs + 7) / 8), dim3(32, 8), 0, stream>>>(
          bufH, attS, attD, aS, aD, Nn, Hh, Cc);
    }
    // init accumulators
    const long long mn = (long long)Nn * Hh;
    fill_kernel<<<1024, 256, 0, stream>>>(mM, -INFINITY, mn);
    fill_kernel<<<1024, 256, 0, stream>>>(den, 0.f, mn);
    fill_kernel<<<2048, 256, 0, stream>>>(bufO, 0.f, (long long)Nn * Nout);
    // dst-grouped softmax over edges (+ self loops)
    const long long et = (long long)(E + Nn) * Hh;
    edge_max<<<dim3((unsigned)((et + 255) / 256)), 256, 0, stream>>>(
        ei, aS, aD, mM, E, Nn, Hh);
    edge_expsum<<<dim3((unsigned)((et + 255) / 256)), 256, 0, stream>>>(
        ei, aS, aD, mM, den, E, Nn, Hh);
    {
      const int edges = E + Nn;
      edge_agg<<<dim3((edges + 7) / 8), dim3(32, 8), 0, stream>>>(
          ei, aS, aD, mM, den, bufH, bufO, E, Nn, Hh, Cc);
    }
    // concat=False -> mean over heads, + bias, relu
    const long long nc = (long long)Nn * Cc;
    head_mean_relu<<<dim3((unsigned)((nc + 255) / 256)), 256, 0, stream>>>(
        bufO, bias, Xout, Nn, Hh, Cc);
  };

  run_layer(x,    IN, W1, as1, ad1, b1, H1h, bufX);
  run_layer(bufX, Cc, W2, as2, ad2, b2, H2h, bufX);  // in-place safe: Xin only
                                                     // read by the first kernel

  // global max pool per graph + linear head
  fill_kernel<<<64, 256, 0, stream>>>(pool, -INFINITY, (long long)Gg * Cc);
  const long long nc = (long long)Nn * Cc;
  pool_max<<<dim3((unsigned)((nc + 255) / 256)), 256, 0, stream>>>(
      bufX, bat, pool, Nn, Cc);
  final_linear<<<dim3((Gg + 7) / 8), dim3(32, 8), 0, stream>>>(
      pool, linW, linb, out, Gg, Cc);
}